// CellAttentionLayer_79207786873545
// MI455X (gfx1250) — compile-verified
//
#include <hip/hip_runtime.h>
#include <hip/hip_bf16.h>

typedef __attribute__((ext_vector_type(2))) float v2f;
typedef __attribute__((ext_vector_type(8))) float v8f;

#define F 32            // F_IN == F_OUT == 32
#define NEG_SLOPE 0.2f
#define MAXBLOCKS 2048

// ---------------------------------------------------------------------------
// Zero-fill a contiguous float region (accumulators + sums) each call.
// ---------------------------------------------------------------------------
__global__ void zero_kernel(float* __restrict__ p, long long n) {
    long long i = (long long)blockIdx.x * blockDim.x + threadIdx.x;
    long long stride = (long long)gridDim.x * blockDim.x;
    for (; i < n; i += stride) p[i] = 0.0f;
}

// ---------------------------------------------------------------------------
// h = x @ W  via V_WMMA_F32_16X16X4_F32. One wave per 16x16 output tile,
// K=32 accumulated as 8 chained K=4 WMMAs. Layouts per CDNA5 ISA 7.12.2:
//   A 16x4 : lanes 0-15 hold (K=0,K=1), lanes 16-31 hold (K=2,K=3), M=lane%16
//   B 4x16 : lanes 0-15 hold (K=0,K=1), lanes 16-31 hold (K=2,K=3), N=lane%16
//   C 16x16: vgpr v -> M = v + (lane>=16 ? 8 : 0), N = lane%16
// ---------------------------------------------------------------------------
__global__ void gemm_wmma_kernel(const float* __restrict__ x,
                                 const float* __restrict__ W,
                                 float* __restrict__ h, int nrows) {
    const int wave  = (blockIdx.x * blockDim.x + threadIdx.x) >> 5;
    const int lane  = threadIdx.x & 31;
    const int tiles_n = F / 16;                 // 2
    const int tile_m = wave / tiles_n;
    const int tile_n = wave % tiles_n;
    if (tile_m * 16 >= nrows) return;           // wave-uniform guard

    const int mrow  = tile_m * 16 + (lane & 15);
    const int col   = tile_n * 16 + (lane & 15);
    const int khalf = (lane >> 4) * 2;          // 0 or 2

    const float* xrow = x + (long long)mrow * F;
    v8f c = {};
#pragma unroll
    for (int kc = 0; kc < 8; ++kc) {
        const int k0 = kc * 4 + khalf;
        v2f a = *(const v2f*)(xrow + k0);       // x[mrow, k0], x[mrow, k0+1]
        v2f b;
        b.x = W[k0 * F + col];
        b.y = W[(k0 + 1) * F + col];
        c = __builtin_amdgcn_wmma_f32_16x16x4_f32(
                /*neg_a=*/false, a, /*neg_b=*/false, b,
                /*c_mod=*/(short)0, c, /*reuse_a=*/false, /*reuse_b=*/false);
    }
    const int mbase = tile_m * 16 + (lane >> 4) * 8;
#pragma unroll
    for (int v = 0; v < 8; ++v)
        h[(long long)(mbase + v) * F + col] = c[v];
}

// ---------------------------------------------------------------------------
// Per-node attention scalars: a_src[n] = h[n]·att[0:32], a_tgt[n] = h[n]·att[32:64]
// One wave per node, lane = feature, wave-shuffle reduce.
// ---------------------------------------------------------------------------
__global__ void alpha_kernel(const float* __restrict__ h,
                             const float* __restrict__ att,
                             float* __restrict__ a_src,
                             float* __restrict__ a_tgt, int n) {
    const int wave   = (blockIdx.x * blockDim.x + threadIdx.x) >> 5;
    const int lane   = threadIdx.x & 31;
    const int nwaves = (gridDim.x * blockDim.x) >> 5;
    const float wa = att[lane];
    const float wb = att[F + lane];
    for (int node = wave; node < n; node += nwaves) {
        float hv = h[(long long)node * F + lane];
        float s = hv * wa;
        float t = hv * wb;
#pragma unroll
        for (int off = 16; off > 0; off >>= 1) {
            s += __shfl_down(s, off, 32);
            t += __shfl_down(t, off, 32);
        }
        if (lane == 0) { a_src[node] = s; a_tgt[node] = t; }
    }
}

__device__ __forceinline__ float lrelu(float v) {
    return v > 0.0f ? v : NEG_SLOPE * v;
}

// ---------------------------------------------------------------------------
// Stage 1 of deterministic global max over leaky_relu(edge scores).
// ---------------------------------------------------------------------------
__global__ void edge_max_kernel(const int* __restrict__ src,
                                const int* __restrict__ tgt,
                                const float* __restrict__ a_src,
                                const float* __restrict__ a_tgt,
                                float* __restrict__ partials, int E) {
    __shared__ float smax[256];
    float m = -3.402823466e38f;
    for (int e = blockIdx.x * blockDim.x + threadIdx.x; e < E;
         e += gridDim.x * blockDim.x) {
        float v = lrelu(a_src[src[e]] + a_tgt[tgt[e]]);
        m = fmaxf(m, v);
    }
    smax[threadIdx.x] = m;
    __syncthreads();
    for (int s = 128; s > 0; s >>= 1) {
        if (threadIdx.x < s)
            smax[threadIdx.x] = fmaxf(smax[threadIdx.x], smax[threadIdx.x + s]);
        __syncthreads();
    }
    if (threadIdx.x == 0) partials[blockIdx.x] = smax[0];
}

__global__ void reduce_max_kernel(const float* __restrict__ partials, int n,
                                  float* __restrict__ out) {
    __shared__ float smax[256];
    float m = -3.402823466e38f;
    for (int i = threadIdx.x; i < n; i += 256) m = fmaxf(m, partials[i]);
    smax[threadIdx.x] = m;
    __syncthreads();
    for (int s = 128; s > 0; s >>= 1) {
        if (threadIdx.x < s)
            smax[threadIdx.x] = fmaxf(smax[threadIdx.x], smax[threadIdx.x + s]);
        __syncthreads();
    }
    if (threadIdx.x == 0) *out = smax[0];
}

// ---------------------------------------------------------------------------
// Edge scatter: one wave per edge, lane = feature.
//   w = exp(lrelu(a_src[s]+a_tgt[t]) - mx)
//   acc[s,:]  += w * h[t,:]   (32 coalesced f32 atomics / edge)
//   sums[s]   += w            (lane 0)
// ---------------------------------------------------------------------------
__global__ void edge_scatter_kernel(const int* __restrict__ src,
                                    const int* __restrict__ tgt,
                                    const float* __restrict__ a_src,
                                    const float* __restrict__ a_tgt,
                                    const float* __restrict__ h,
                                    const float* __restrict__ mx,
                                    float* __restrict__ acc,
                                    float* __restrict__ sums, int E) {
    const int wave   = (blockIdx.x * blockDim.x + threadIdx.x) >> 5;
    const int lane   = threadIdx.x & 31;
    const int nwaves = (gridDim.x * blockDim.x) >> 5;
    const float m = *mx;
    for (int e = wave; e < E; e += nwaves) {
        const int s = src[e];
        const int t = tgt[e];
        const float w = __expf(lrelu(a_src[s] + a_tgt[t]) - m);
        atomicAdd(&acc[(long long)s * F + lane], w * h[(long long)t * F + lane]);
        if (lane == 0) atomicAdd(&sums[s], w);
    }
}

// ---------------------------------------------------------------------------
// out = elu( acc_irr/(sums_irr+1e-10) + acc_sol/(sums_sol+1e-10) )
// ---------------------------------------------------------------------------
__global__ void finalize_kernel(const float* __restrict__ acc_i,
                                const float* __restrict__ sums_i,
                                const float* __restrict__ acc_s,
                                const float* __restrict__ sums_s,
                                float* __restrict__ out, int n) {
    int i = blockIdx.x * blockDim.x + threadIdx.x;
    if (i >= n) return;
    int node = i >> 5;  // /F
    float v = acc_i[i] / (sums_i[node] + 1e-10f) +
              acc_s[i] / (sums_s[node] + 1e-10f);
    out[i] = v > 0.0f ? v : (__expf(v) - 1.0f);
}

extern "C" void kernel_launch(void* const* d_in, const int* in_sizes, int n_in,
                              void* d_out, int out_size, void* d_ws, size_t ws_size,
                              hipStream_t stream) {
    const float* x        = (const float*)d_in[0];
    const int*   do_index = (const int*)d_in[1];
    const int*   up_index = (const int*)d_in[2];
    const float* Wirr     = (const float*)d_in[3];
    const float* Wsol     = (const float*)d_in[4];
    const float* att_irr  = (const float*)d_in[5];
    const float* att_sol  = (const float*)d_in[6];
    float* out = (float*)d_out;

    const int N = in_sizes[0] / F;      // 100000
    const int E = in_sizes[1] / 2;      // 1600000

    // do/up index layout: shape (2, E) row-major -> [0:E)=src, [E:2E)=tgt
    const int* do_src = do_index;       const int* do_tgt = do_index + E;
    const int* up_src = up_index;       const int* up_tgt = up_index + E;

    // ---- workspace carve-up (floats) ----
    float* ws = (float*)d_ws;
    long long off = 0;
    float* h_irr    = ws + off; off += (long long)N * F;
    float* h_sol    = ws + off; off += (long long)N * F;
    float* acc_irr  = ws + off; off += (long long)N * F;   // zeroed
    float* acc_sol  = ws + off; off += (long long)N * F;   // zeroed
    float* sums_irr = ws + off; off += N;                  // zeroed
    float* sums_sol = ws + off; off += N;                  // zeroed
    float* asrc_irr = ws + off; off += N;
    float* atgt_irr = ws + off; off += N;
    float* asrc_sol = ws + off; off += N;
    float* atgt_sol = ws + off; off += N;
    float* part_irr = ws + off; off += MAXBLOCKS;
    float* part_sol = ws + off; off += MAXBLOCKS;
    float* mx_irr   = ws + off; off += 1;
    float* mx_sol   = ws + off; off += 1;

    const long long zero_n = 2LL * N * F + 2LL * N;   // acc_irr..sums_sol contiguous
    zero_kernel<<<2048, 256, 0, stream>>>(acc_irr, zero_n);

    // ---- GEMMs via WMMA: one wave per 16x16 tile ----
    const int tiles_m = (N + 15) / 16;
    const int waves_needed = tiles_m * (F / 16);
    const int gemm_blocks = (waves_needed + 7) / 8;   // 8 waves / 256-thread block
    gemm_wmma_kernel<<<gemm_blocks, 256, 0, stream>>>(x, Wirr, h_irr, N);
    gemm_wmma_kernel<<<gemm_blocks, 256, 0, stream>>>(x, Wsol, h_sol, N);

    // ---- per-node attention scalars ----
    alpha_kernel<<<512, 256, 0, stream>>>(h_irr, att_irr, asrc_irr, atgt_irr, N);
    alpha_kernel<<<512, 256, 0, stream>>>(h_sol, att_sol, asrc_sol, atgt_sol, N);

    // ---- global max per branch (2-stage, deterministic) ----
    edge_max_kernel<<<MAXBLOCKS, 256, 0, stream>>>(do_src, do_tgt, asrc_irr, atgt_irr, part_irr, E);
    edge_max_kernel<<<MAXBLOCKS, 256, 0, stream>>>(up_src, up_tgt, asrc_sol, atgt_sol, part_sol, E);
    reduce_max_kernel<<<1, 256, 0, stream>>>(part_irr, MAXBLOCKS, mx_irr);
    reduce_max_kernel<<<1, 256, 0, stream>>>(part_sol, MAXBLOCKS, mx_sol);

    // ---- edge scatter (one wave per edge, lane = feature) ----
    edge_scatter_kernel<<<MAXBLOCKS, 256, 0, stream>>>(do_src, do_tgt, asrc_irr, atgt_irr,
                                                       h_irr, mx_irr, acc_irr, sums_irr, E);
    edge_scatter_kernel<<<MAXBLOCKS, 256, 0, stream>>>(up_src, up_tgt, asrc_sol, atgt_sol,
                                                       h_sol, mx_sol, acc_sol, sums_sol, E);

    // ---- finalize: divide by softmax sums, add branches, ELU ----
    const int total = N * F;
    finalize_kernel<<<(total + 255) / 256, 256, 0, stream>>>(
        acc_irr, sums_irr, acc_sol, sums_sol, out, total);
}